// VQ_20761871909708
// MI455X (gfx1250) — compile-verified
//
#include <hip/hip_runtime.h>

typedef __attribute__((ext_vector_type(2))) float v2f;
typedef __attribute__((ext_vector_type(8))) float v8f;

// ---------------------------------------------------------------------------
// Kernel 1: per-code squared norms  csq[k] = sum_d CB[k][d]^2   (K x 64 f32)
// ---------------------------------------------------------------------------
__global__ void vq_csq_kernel(const float* __restrict__ cb,
                              float* __restrict__ csq, int K) {
  int k = blockIdx.x * blockDim.x + threadIdx.x;
  if (k >= K) return;
  const float4* row = reinterpret_cast<const float4*>(cb + (size_t)k * 64);
  float s = 0.f;
#pragma unroll
  for (int i = 0; i < 16; ++i) {
    float4 v = row[i];
    s = fmaf(v.x, v.x, s);
    s = fmaf(v.y, v.y, s);
    s = fmaf(v.z, v.z, s);
    s = fmaf(v.w, v.w, s);
  }
  csq[k] = s;
}

// ---------------------------------------------------------------------------
// Kernel 2: fused  scores = -2*X@CB^T + csq  -> argmin over K -> gather rows.
// One wave32 handles a 16-row strip of X and sweeps all K codes in 16-column
// chunks using V_WMMA_F32_16X16X4_F32 (f32 in / f32 accumulate).
// ---------------------------------------------------------------------------
__global__ __launch_bounds__(256) void vq_argmin_kernel(
    const float* __restrict__ X, const float* __restrict__ CB,
    const float* __restrict__ csq, float* __restrict__ out, int N, int K) {
  const int lane = threadIdx.x & 31;
  const int half = lane >> 4;  // 0: lanes 0-15, 1: lanes 16-31
  const int nl   = lane & 15;
  const int wave = blockIdx.x * (blockDim.x >> 5) + (threadIdx.x >> 5);
  const int m0   = wave * 16;
  if (m0 >= N) return;  // wave-uniform: EXEC stays all-ones for WMMA

  // Preload A strip (16 rows x 64 cols) into 16 WMMA 16x4 f32 A-fragments.
  // A layout: lanes 0-15 hold row=lane, K={kb,kb+1}; lanes 16-31 K={kb+2,kb+3}.
  v2f a[16];
  {
    const float* xrow = X + (size_t)(m0 + nl) * 64;
#pragma unroll
    for (int kk = 0; kk < 16; ++kk) {
      const float2 v =
          *reinterpret_cast<const float2*>(xrow + kk * 4 + half * 2);
      a[kk].x = v.x;
      a[kk].y = v.y;
    }
  }

  float best[8];
  int   bidx[8];
#pragma unroll
  for (int e = 0; e < 8; ++e) {
    best[e] = 3.4e38f;
    bidx[e] = 0;
  }

  for (int n0 = 0; n0 < K; n0 += 16) {
    // gfx1250 global_prefetch_b8 on the next codebook chunk (L2-resident).
    if (n0 + 16 < K)
      __builtin_prefetch(CB + (size_t)(n0 + 16 + nl) * 64, 0, 0);

    // B fragment: B[k][n] = CB[n0+n][kb+k]; column n = lane&15 for all k,
    // mirroring the A layout across half-waves.
    const float* crow = CB + (size_t)(n0 + nl) * 64;
    v2f b[16];
#pragma unroll
    for (int kk = 0; kk < 16; ++kk) {
      const float2 v =
          *reinterpret_cast<const float2*>(crow + kk * 4 + half * 2);
      b[kk].x = v.x;
      b[kk].y = v.y;
    }
    const float cs = csq[n0 + nl];

    v8f acc = {0.f, 0.f, 0.f, 0.f, 0.f, 0.f, 0.f, 0.f};
#pragma unroll
    for (int kk = 0; kk < 16; ++kk) {
      acc = __builtin_amdgcn_wmma_f32_16x16x4_f32(
          /*neg_a=*/false, a[kk], /*neg_b=*/false, b[kk],
          /*c_mod=*/(short)0, acc, /*reuse_a=*/false, /*reuse_b=*/false);
    }

    // score = csq[n] - 2 * (x . c);  column n == lane&15 for every acc VGPR.
#pragma unroll
    for (int e = 0; e < 8; ++e) {
      float s = fmaf(-2.f, acc[e], cs);
      if (s < best[e]) {
        best[e] = s;
        bidx[e] = n0 + nl;
      }
    }
  }

  // Min+argmin across the 16 lanes of each half (offsets 8..1 stay in-half).
#pragma unroll
  for (int off = 8; off >= 1; off >>= 1) {
#pragma unroll
    for (int e = 0; e < 8; ++e) {
      float os = __shfl_xor(best[e], off, 32);
      int   oi = __shfl_xor(bidx[e], off, 32);
      if (os < best[e] || (os == best[e] && oi < bidx[e])) {
        best[e] = os;
        bidx[e] = oi;
      }
    }
  }

  // Gather winning codebook rows, write 16 output rows (float2 per lane).
#pragma unroll
  for (int j = 0; j < 16; ++j) {
    const int idx = __shfl(bidx[j & 7], (j >> 3) << 4, 32);
    const float2 v =
        *reinterpret_cast<const float2*>(CB + (size_t)idx * 64 + lane * 2);
    *reinterpret_cast<float2*>(out + (size_t)(m0 + j) * 64 + lane * 2) = v;
  }
}

// ---------------------------------------------------------------------------
extern "C" void kernel_launch(void* const* d_in, const int* in_sizes, int n_in,
                              void* d_out, int out_size, void* d_ws,
                              size_t ws_size, hipStream_t stream) {
  const float* X  = (const float*)d_in[0];  // encodings (16,32,32,64) f32
  const float* CB = (const float*)d_in[1];  // codebook  (8192,64)     f32
  float* out = (float*)d_out;

  const int D = 64;
  const int N = in_sizes[0] / D;  // 16384
  const int K = in_sizes[1] / D;  // 8192
  float* csq = (float*)d_ws;      // K floats of scratch

  vq_csq_kernel<<<(K + 255) / 256, 256, 0, stream>>>(CB, csq, K);

  const int waves  = (N + 15) / 16;
  const int blocks = (waves + 7) / 8;  // 8 waves (256 threads) per block
  vq_argmin_kernel<<<blocks, 256, 0, stream>>>(X, CB, csq, out, N, K);
}